// FragmentPositionDistribution2_36292473651627
// MI455X (gfx1250) — compile-verified
//
#include <hip/hip_runtime.h>
#include <math.h>

#define BLOCK 128

// ---- CDNA5 async global->LDS staging (ASYNCcnt-tracked DMA path) ----------
// Each lane copies 16B per instruction from global address gaddr + K*16 to LDS address lds + K*16.
// NT hint: the 448MB bincount stream is read-once -> don't thrash 192MB L2.
template <int K, int KN>
struct AsyncCopy {
    static __device__ __forceinline__ void run(unsigned lds, unsigned long long gaddr) {
        asm volatile("global_load_async_to_lds_b128 %0, %1, off offset:%2 th:TH_LOAD_NT"
                     :: "v"(lds), "v"(gaddr), "n"(K * 16)
                     : "memory");
        AsyncCopy<K + 1, KN>::run(lds, gaddr);
    }
};
template <int KN>
struct AsyncCopy<KN, KN> {
    static __device__ __forceinline__ void run(unsigned, unsigned long long) {}
};

// ---- per-region log-softmax pick: two passes over the LDS-staged row ------
// uh[s] = (bc[s] > 1 ? d : 0) + base[s];  return uh[bx] - logsumexp(uh) + log(S)
// Target entry fetched directly (bx < 16 < S always, per reference) instead of
// a 2*S-op v_cmp/v_cndmask chain.
template <int NV4>
__device__ __forceinline__ float region_term(const int4* __restrict__ row,
                                             const float* __restrict__ base,
                                             float dval, int bx) {
    const float4* __restrict__ b4 = (const float4*)base;
    const int*    __restrict__ ri = (const int*)row;

    // pass 1: row max (ds_load_b128 + float4 from L2-hot baseline)
    float m = -INFINITY;
#pragma unroll
    for (int v = 0; v < NV4; ++v) {
        int4   b = row[v];
        float4 w = b4[v];
        float u0 = (b.x > 1 ? dval : 0.0f) + w.x;
        float u1 = (b.y > 1 ? dval : 0.0f) + w.y;
        float u2 = (b.z > 1 ? dval : 0.0f) + w.z;
        float u3 = (b.w > 1 ? dval : 0.0f) + w.w;
        m = fmaxf(m, fmaxf(fmaxf(u0, u1), fmaxf(u2, u3)));
    }

    // pass 2: sum of exp
    float sum = 0.0f;
#pragma unroll
    for (int v = 0; v < NV4; ++v) {
        int4   b = row[v];
        float4 w = b4[v];
        float u0 = (b.x > 1 ? dval : 0.0f) + w.x;
        float u1 = (b.y > 1 ? dval : 0.0f) + w.y;
        float u2 = (b.z > 1 ? dval : 0.0f) + w.z;
        float u3 = (b.w > 1 ? dval : 0.0f) + w.w;
        sum += __expf(u0 - m) + __expf(u1 - m) + __expf(u2 - m) + __expf(u3 - m);
    }

    // direct target fetch: 1 ds_load_b32 + 1 global_load_b32
    const float tv = (ri[bx] > 1 ? dval : 0.0f) + base[bx];
    return tv - m - __logf(sum) + logf((float)(NV4 * 4));
}

__global__ __launch_bounds__(BLOCK)
void FragmentPositionDistribution2_36292473651627_kernel(
    const int* __restrict__ bc0, const int* __restrict__ bc1, const int* __restrict__ bc2,
    const int* __restrict__ gbin, const int* __restrict__ binx,
    const int* __restrict__ labels, const int* __restrict__ cellix,
    const float* __restrict__ base0, const float* __restrict__ diff0,
    const float* __restrict__ base1, const float* __restrict__ diff1,
    const float* __restrict__ base2, const float* __restrict__ diff2,
    float* __restrict__ out, int n)
{
    // Padded LDS rows: strides 17/9/5 int4 -> bank(t,s) = (4*P*t+s) mod 64 (P = stride),
    // 2-way overlap == the 512B/wave b128 floor. 16B alignment guaranteed.
    __shared__ int4 sm0[BLOCK * 17];  // S=64: 256B data + 16B pad per row
    __shared__ int4 sm1[BLOCK * 9];   // S=32: 128B data + 16B pad
    __shared__ int4 sm2[BLOCK * 5];   // S=16:  64B data + 16B pad

    const int t    = threadIdx.x;
    const int frag = blockIdx.x * BLOCK + t;
    if (frag >= n) return;

    // low 32 bits of the generic pointer == LDS byte offset
    const unsigned l0 = (unsigned)(unsigned long long)(void*)&sm0[t * 17];
    const unsigned l1 = (unsigned)(unsigned long long)(void*)&sm1[t * 9];
    const unsigned l2 = (unsigned)(unsigned long long)(void*)&sm2[t * 5];

    // kick off the DMA of this fragment's three bincount rows (448B/lane).
    // Issue order matters: ASYNCcnt decrements in order, enabling the
    // staged waits below (16, then 8, then 4 copies).
    AsyncCopy<0, 16>::run(l0, (unsigned long long)(size_t)(bc0 + (size_t)frag * 64));
    AsyncCopy<0,  8>::run(l1, (unsigned long long)(size_t)(bc1 + (size_t)frag * 32));
    AsyncCopy<0,  4>::run(l2, (unsigned long long)(size_t)(bc2 + (size_t)frag * 16));

    // small gathers while the async copies are in flight
    const int lab = labels[cellix[frag]];
    const int gb0 = gbin[frag * 3 + 0];
    const int gb1 = gbin[frag * 3 + 1];
    const int gb2 = gbin[frag * 3 + 2];
    const int bx0 = binx[frag * 3 + 0];
    const int bx1 = binx[frag * 3 + 1];
    const int bx2 = binx[frag * 3 + 2];

    const float* b0p = base0 + (size_t)gb0 * 64;
    const float* b1p = base1 + (size_t)gb1 * 32;
    const float* b2p = base2 + (size_t)gb2 * 16;
    __builtin_prefetch(b0p, 0, 3);   // global_prefetch_b8 — pull baseline rows
    __builtin_prefetch(b1p, 0, 3);
    __builtin_prefetch(b2p, 0, 3);

    const float dv0 = diff0[lab];
    const float dv1 = diff1[lab];
    const float dv2 = diff2[lab];

    float lp = 0.0f;

    // staged waits: only this wave consumes this wave's DMA -> no barrier.
    // 28 ops in flight; first 16 done when cnt<=12, first 24 when cnt<=4.
    asm volatile("s_wait_asynccnt 0xc" ::: "memory");
    lp += region_term<16>(&sm0[t * 17], b0p, dv0, bx0);

    asm volatile("s_wait_asynccnt 0x4" ::: "memory");
    lp += region_term<8 >(&sm1[t * 9 ], b1p, dv1, bx1);

    asm volatile("s_wait_asynccnt 0x0" ::: "memory");
    lp += region_term<4 >(&sm2[t * 5 ], b2p, dv2, bx2);

    out[frag] = lp;
}

extern "C" void kernel_launch(void* const* d_in, const int* in_sizes, int n_in,
                              void* d_out, int out_size, void* d_ws, size_t ws_size,
                              hipStream_t stream) {
    // setup_inputs() dict order:
    // 0..2: bincounts_0/1/2, 3: global_binixs, 4: binixs, 5: labels,
    // 6: local_cell_ix, 7: baseline_0, 8: diff_0, 9: baseline_1, 10: diff_1,
    // 11: baseline_2, 12: diff_2
    const int*   bc0    = (const int*)  d_in[0];
    const int*   bc1    = (const int*)  d_in[1];
    const int*   bc2    = (const int*)  d_in[2];
    const int*   gbin   = (const int*)  d_in[3];
    const int*   binx   = (const int*)  d_in[4];
    const int*   labels = (const int*)  d_in[5];
    const int*   cellix = (const int*)  d_in[6];
    const float* base0  = (const float*)d_in[7];
    const float* diff0  = (const float*)d_in[8];
    const float* base1  = (const float*)d_in[9];
    const float* diff1  = (const float*)d_in[10];
    const float* base2  = (const float*)d_in[11];
    const float* diff2  = (const float*)d_in[12];
    float*       out    = (float*)d_out;

    const int n    = in_sizes[6];            // N fragments
    const int grid = (n + BLOCK - 1) / BLOCK;

    FragmentPositionDistribution2_36292473651627_kernel<<<grid, BLOCK, 0, stream>>>(
        bc0, bc1, bc2, gbin, binx, labels, cellix,
        base0, diff0, base1, diff1, base2, diff2, out, n);
}